// VectorQuantizer_12592844112281
// MI455X (gfx1250) — compile-verified
//
#include <hip/hip_runtime.h>
#include <hip/hip_bf16.h>

typedef float v2f __attribute__((ext_vector_type(2)));
typedef float v8f __attribute__((ext_vector_type(8)));
typedef unsigned int u32x4 __attribute__((ext_vector_type(4)));
typedef int i32x4 __attribute__((ext_vector_type(4)));
typedef int i32x8 __attribute__((ext_vector_type(8)));

#define K_CODES 1024
#define DIM     256
#define HWSZ    1024        // 32*32 spatial positions per batch image
#define NROWS   32768       // 32 * 32 * 32

// d_out float offsets: (loss, z_q_out, perplexity, min_encodings, idx)
#define OFF_LOSS   0
#define OFF_ZQ     1
#define OFF_PERP   8388609ll
#define OFF_ONEHOT 8388610ll
#define OFF_IDX    41943042ll

// -------------------- kernel 1: ||e_k||^2 + zero counts --------------------
__global__ void vq_ek2_kernel(const float* __restrict__ emb,
                              float* __restrict__ ek2,
                              int* __restrict__ counts) {
    __shared__ float sh[256];
    const int k = blockIdx.x;
    const float v = emb[k * DIM + threadIdx.x];
    sh[threadIdx.x] = v * v;
    __syncthreads();
    for (int s = 128; s > 0; s >>= 1) {
        if (threadIdx.x < (unsigned)s) sh[threadIdx.x] += sh[threadIdx.x + s];
        __syncthreads();
    }
    if (threadIdx.x == 0) { ek2[k] = sh[0]; counts[k] = 0; }
}

// -------------------- kernel 2: fused WMMA distance + argmin --------------------
// Block = 128 threads (4 waves). Block handles 64 rows of zf; each wave a 16-row strip.
// z tile staged by the Tensor Data Mover: LDS layout ldsz[d*64 + m] (64 KB).
__global__ __launch_bounds__(128) void vq_argmin_kernel(
    const float* __restrict__ z,     // [32, 256, 32, 32]
    const float* __restrict__ emb,   // [1024, 256]
    const float* __restrict__ ek2,   // [1024]
    int*  __restrict__ idxbuf,       // [32768]
    int*  __restrict__ counts,       // [1024]
    float* __restrict__ idx_out) {   // [32768] as float
    extern __shared__ float ldsz[];  // 16384 floats = 64 KB

    const int tid       = threadIdx.x;
    const int wid       = tid >> 5;
    const int rows_base = blockIdx.x * 64;
    const int b         = rows_base >> 10;
    const int hw_base   = rows_base & (HWSZ - 1);   // multiple of 64
    const float* zb     = z + (size_t)b * (DIM * HWSZ);

    // ---- TDM staging: each wave DMAs a 64-dim slab (64 lines x 256B) into LDS ----
    {
        const int dbase = wid * 64;
        const unsigned long long ga =
            reinterpret_cast<unsigned long long>(zb + (size_t)dbase * HWSZ + hw_base);
        const unsigned lds_addr =
            (unsigned)reinterpret_cast<unsigned long long>(ldsz) + (unsigned)(dbase * 64 * 4);

        u32x4 g0;
        g0[0] = 1u;                                            // count=1, user mode
        g0[1] = lds_addr;                                      // LDS byte address
        g0[2] = (unsigned)(ga & 0xffffffffull);                // global_addr[31:0]
        g0[3] = (unsigned)((ga >> 32) & 0x1ffffffull)          // global_addr[56:32]
              | (2u << 30);                                    // type = 2 ("image")

        i32x8 g1;
        g1[0] = (2 << 16);            // workgroup_mask=0, data_size=2 (4B), no flags
        g1[1] = (int)(1024u << 16);   // tensor_dim0 = 1024 (low16 in [31:16])
        g1[2] = (int)(256u << 16);    // tensor_dim0 hi=0, tensor_dim1 = 256
        g1[3] = (int)(64u << 16);     // tensor_dim1 hi=0, tile_dim0 = 64 (contig elems)
        g1[4] = 64;                   // tile_dim1 = 64 lines, tile_dim2 = 0 (2D)
        g1[5] = 1024;                 // tensor_dim0_stride[31:0] = 1024 elements
        g1[6] = 0;                    // stride0 hi, tensor_dim1_stride lo (unused, 2D)
        g1[7] = 0;

        const i32x4 gz4 = {0, 0, 0, 0};
        const i32x8 gz8 = {0, 0, 0, 0, 0, 0, 0, 0};
        __builtin_amdgcn_tensor_load_to_lds(g0, g1, gz4, gz4, gz8, 0);
        __builtin_amdgcn_s_wait_tensorcnt(0);
    }
    __syncthreads();

    const int lane   = tid & 31;
    const int half   = lane >> 4;      // selects K pair within 16x16x4 chunk
    const int lane16 = lane & 15;
    const int mrow   = wid * 16 + lane16;

    float minval[8];
    int   minidx[8];
    #pragma unroll
    for (int v = 0; v < 8; ++v) { minval[v] = 3.4e38f; minidx[v] = 0; }

    const v8f vzero = {0.f, 0.f, 0.f, 0.f, 0.f, 0.f, 0.f, 0.f};

    for (int ctg = 0; ctg < 16; ++ctg) {         // groups of 4 code tiles (64 codes)
        const int c0 = ctg * 64 + lane16;
        const float* e0 = emb + (size_t)c0 * DIM + half * 2;
        v8f acc[4];
        #pragma unroll
        for (int t = 0; t < 4; ++t) acc[t] = vzero;

        #pragma unroll 2
        for (int kc = 0; kc < 64; ++kc) {        // K chunks of 4 over D=256
            const int d0 = kc * 4 + half * 2;
            v2f a;                               // A[m][d0], A[m][d0+1]
            a.x = ldsz[d0 * 64 + mrow];
            a.y = ldsz[d0 * 64 + mrow + 64];
            #pragma unroll
            for (int t = 0; t < 4; ++t) {        // 4 independent accumulation chains
                v2f bb = *reinterpret_cast<const v2f*>(e0 + t * (16 * DIM) + kc * 4);
                acc[t] = __builtin_amdgcn_wmma_f32_16x16x4_f32(
                    false, a, false, bb, (short)0, acc[t], false, false);
            }
        }

        #pragma unroll
        for (int t = 0; t < 4; ++t) {
            const int code = c0 + t * 16;
            const float e2 = ek2[code];
            #pragma unroll
            for (int v = 0; v < 8; ++v) {
                // dist = ||e||^2 - 2 z.e  (||z||^2 is row-constant: argmin-invariant)
                const float dist = fmaf(-2.0f, acc[t][v], e2);
                if (dist < minval[v]) { minval[v] = dist; minidx[v] = code; }
            }
        }
    }

    // Reduce across the 16 lanes of each half-wave (xor stays within half).
    #pragma unroll
    for (int off = 1; off < 16; off <<= 1) {
        #pragma unroll
        for (int v = 0; v < 8; ++v) {
            const float ov = __shfl_xor(minval[v], off, 32);
            const int   oi = __shfl_xor(minidx[v], off, 32);
            if (ov < minval[v] || (ov == minval[v] && oi < minidx[v])) {
                minval[v] = ov; minidx[v] = oi;
            }
        }
    }

    if (lane16 == 0) {   // lane 0 -> rows 0..7 of strip; lane 16 -> rows 8..15
        #pragma unroll
        for (int v = 0; v < 8; ++v) {
            const int n = rows_base + wid * 16 + half * 8 + v;
            const int c = minidx[v];
            idxbuf[n]  = c;
            idx_out[n] = (float)c;
            atomicAdd(&counts[c], 1);
        }
    }
}

// -------------------- kernel 3: one-hot encodings (128 MB stream) --------------------
__global__ void vq_onehot_kernel(const int* __restrict__ idxbuf,
                                 float* __restrict__ onehot) {
    const int n    = blockIdx.x;
    const int code = idxbuf[n];
    float* row = onehot + (size_t)n * K_CODES;
    const int k0 = threadIdx.x * 4;
    float2 a, c;
    a.x = (k0     == code) ? 1.0f : 0.0f;
    a.y = (k0 + 1 == code) ? 1.0f : 0.0f;
    c.x = (k0 + 2 == code) ? 1.0f : 0.0f;
    c.y = (k0 + 3 == code) ? 1.0f : 0.0f;
    *reinterpret_cast<float2*>(row + k0)     = a;   // row base is 8B-aligned
    *reinterpret_cast<float2*>(row + k0 + 2) = c;
}

// -------------------- kernel 4: z_q gather + straight-through + loss partials ----
__global__ void vq_zq_kernel(const float* __restrict__ z,
                             const float* __restrict__ emb,
                             const int* __restrict__ idxbuf,
                             float* __restrict__ zq_out,
                             float* __restrict__ partials) {
    __shared__ float sh[256];
    const int bd = blockIdx.x;           // b*256 + d
    const int b  = bd >> 8;
    const int d  = bd & 255;
    const float* zp = z      + (size_t)b * (DIM * HWSZ) + (size_t)d * HWSZ;
    float*       op = zq_out + (size_t)b * (DIM * HWSZ) + (size_t)d * HWSZ;
    const int*   ip = idxbuf + b * HWSZ;

    float acc = 0.f;
    #pragma unroll
    for (int j = 0; j < 4; ++j) {
        const int hw   = j * 256 + threadIdx.x;
        const int code = ip[hw];
        const float zq = emb[code * DIM + d];
        const float zv = zp[hw];
        const float t  = zq - zv;
        op[hw] = zv + t;                 // z + sg(z_q - z): same fp arithmetic as reference
        acc += t * t;
    }
    sh[threadIdx.x] = acc;
    __syncthreads();
    for (int s = 128; s > 0; s >>= 1) {
        if (threadIdx.x < (unsigned)s) sh[threadIdx.x] += sh[threadIdx.x + s];
        __syncthreads();
    }
    if (threadIdx.x == 0) partials[bd] = sh[0];
}

// -------------------- kernel 5: deterministic final reductions --------------------
__global__ void vq_final_kernel(const float* __restrict__ partials,  // [8192]
                                const int* __restrict__ counts,      // [1024]
                                float* __restrict__ out_loss,
                                float* __restrict__ out_perp) {
    __shared__ float sh[256];
    const int t = threadIdx.x;

    float s = 0.f;
    for (int i = t; i < 8192; i += 256) s += partials[i];  // fixed order
    sh[t] = s;
    __syncthreads();
    for (int k = 128; k > 0; k >>= 1) {
        if (t < k) sh[t] += sh[t + k];
        __syncthreads();
    }
    if (t == 0) *out_loss = 1.25f * (sh[0] / 8388608.0f);  // (1+beta) * MSE
    __syncthreads();

    float h = 0.f;
    for (int i = t; i < 1024; i += 256) {
        const float em = (float)counts[i] / 32768.0f;
        h -= em * logf(em + 1e-10f);
    }
    __syncthreads();
    sh[t] = h;
    __syncthreads();
    for (int k = 128; k > 0; k >>= 1) {
        if (t < k) sh[t] += sh[t + k];
        __syncthreads();
    }
    if (t == 0) *out_perp = expf(sh[0]);
}

// -------------------- launcher --------------------
extern "C" void kernel_launch(void* const* d_in, const int* in_sizes, int n_in,
                              void* d_out, int out_size, void* d_ws, size_t ws_size,
                              hipStream_t stream) {
    const float* z   = (const float*)d_in[0];   // [32, 256, 32, 32]
    const float* emb = (const float*)d_in[1];   // [1024, 256]
    float* out = (float*)d_out;

    // workspace: ek2[1024] | counts[1024] | idx[32768] | partials[8192]  (~172 KB)
    float* ek2      = (float*)d_ws;
    int*   counts   = (int*)d_ws + 1024;
    int*   idxbuf   = (int*)d_ws + 2048;
    float* partials = (float*)d_ws + 2048 + 32768;

    vq_ek2_kernel   <<<1024,  256, 0,     stream>>>(emb, ek2, counts);
    vq_argmin_kernel<<<512,   128, 65536, stream>>>(z, emb, ek2, idxbuf, counts,
                                                    out + OFF_IDX);
    vq_onehot_kernel<<<32768, 256, 0,     stream>>>(idxbuf, out + OFF_ONEHOT);
    vq_zq_kernel    <<<8192,  256, 0,     stream>>>(z, emb, idxbuf, out + OFF_ZQ, partials);
    vq_final_kernel <<<1,     256, 0,     stream>>>(partials, counts,
                                                    out + OFF_LOSS, out + OFF_PERP);
}